// GatedODEFlow_19207093748174
// MI455X (gfx1250) — compile-verified
//
#include <hip/hip_runtime.h>

typedef float v2f __attribute__((ext_vector_type(2)));
typedef float v4f __attribute__((ext_vector_type(4)));
typedef float v8f __attribute__((ext_vector_type(8)));

#define D_MODEL 4096
#define K_SUB   64
#define ROWS_PER_WAVE 32                   // two 16-row WMMA tiles per wave
#define WAVES_PER_BLOCK 8
#define ROWS_PER_BLOCK (ROWS_PER_WAVE * WAVES_PER_BLOCK)   // 256
#define COL4 (D_MODEL / 4)                 // 1024

__global__ __launch_bounds__(256) void gated_ode_flow_kernel(
    const float* __restrict__ x,
    const float* __restrict__ mu,
    const float* __restrict__ U,
    const float* __restrict__ S,
    const float* __restrict__ tgt,
    const float* __restrict__ log_step_p,
    const float* __restrict__ sigma_p,
    const int*   __restrict__ num_steps_p,
    float* __restrict__ out)
{
    __shared__ float lds_projmu[K_SUB];
    __shared__ float lds_zt[K_SUB];
    __shared__ float lds_invS[K_SUB];
    __shared__ float lds_P[ROWS_PER_BLOCK];

    const int tid  = threadIdx.x;
    const int lane = tid & 31;
    const int wave = tid >> 5;

    // ---- prologue: fixed 64-vectors projmu = U^T mu, zt = invS .* U^T (t - mu) ----
    if (tid < K_SUB) {
        const int n = tid;
        float pm = 0.f, pt = 0.f;
        for (int d = 0; d < D_MODEL; ++d) {
            const float u = U[(size_t)d * K_SUB + n];
            pm = fmaf(u, mu[d],  pm);
            pt = fmaf(u, tgt[d], pt);
        }
        const float invs = 1.0f / (S[n] + 1e-6f);
        lds_projmu[n] = pm;
        lds_invS[n]   = invs;
        lds_zt[n]     = (pt - pm) * invs;
    }
    __syncthreads();

    // ---- one fp32 WMMA GEMM: proj0 = U^T x  (mu folded into projmu) ----
    const int rsel = lane & 15;            // A: row within 16-tile ; B/C: col within n-tile
    const int kk   = (lane >> 4) << 1;     // A/B K sub-offset: lanes 0-15 -> K 0,1 ; lanes 16-31 -> K 2,3

    const int blockRow = blockIdx.x * ROWS_PER_BLOCK;
    const int waveRow  = blockRow + wave * ROWS_PER_WAVE;

    const float* pA0 = x + (size_t)(waveRow + rsel)      * D_MODEL + kk;
    const float* pA1 = x + (size_t)(waveRow + 16 + rsel) * D_MODEL + kk;
    const float* pB  = U + (size_t)kk * K_SUB + rsel;

    v8f acc[2][4];
    #pragma unroll
    for (int m = 0; m < 2; ++m)
        #pragma unroll
        for (int t = 0; t < 4; ++t)
            acc[m][t] = (v8f){0.f,0.f,0.f,0.f,0.f,0.f,0.f,0.f};

    #pragma unroll 2
    for (int k0 = 0; k0 < D_MODEL; k0 += 4) {
        v2f a0 = *(const v2f*)(pA0 + k0);
        v2f a1 = *(const v2f*)(pA1 + k0);
        const float* b = pB + (size_t)k0 * K_SUB;
        #pragma unroll
        for (int t = 0; t < 4; ++t) {
            v2f bv;
            bv.x = b[t * 16];               // U[k0+kk  ][16t + rsel]
            bv.y = b[t * 16 + K_SUB];       // U[k0+kk+1][16t + rsel]
            acc[0][t] = __builtin_amdgcn_wmma_f32_16x16x4_f32(
                false, a0, false, bv, (short)0, acc[0][t], false, false);
            acc[1][t] = __builtin_amdgcn_wmma_f32_16x16x4_f32(
                false, a1, false, bv, (short)0, acc[1][t], false, false);
        }
    }

    // ---- gate reduction + closed-form 5-step recurrence -> P per row ----
    float pmu_l[4], zt_l[4], is_l[4];
    #pragma unroll
    for (int t = 0; t < 4; ++t) {
        const int n = t * 16 + rsel;
        pmu_l[t] = lds_projmu[n];
        zt_l[t]  = lds_zt[n];
        is_l[t]  = lds_invS[n];
    }

    const float sigma  = *sigma_p;
    const int   nsteps = *num_steps_p;
    float step = __expf(*log_step_p);
    step = fminf(fmaxf(step, 1e-3f), 1.0f);
    const float h = step / (float)nsteps;
    const float inv_den = 1.0f / ((float)K_SUB * 2.0f * sigma * sigma);  // T = 1

    #pragma unroll
    for (int m = 0; m < 2; ++m) {
        #pragma unroll
        for (int r = 0; r < 8; ++r) {
            float q = 0.f, c = 0.f, qt = 0.f;
            #pragma unroll
            for (int t = 0; t < 4; ++t) {
                const float z = (acc[m][t][r] - pmu_l[t]) * is_l[t];
                q  = fmaf(z, z,       q);
                c  = fmaf(z, zt_l[t], c);
                qt = fmaf(zt_l[t], zt_l[t], qt);
            }
            // reduce within each 16-lane half (rows r and r+8 stay separate)
            #pragma unroll
            for (int off = 8; off >= 1; off >>= 1) {
                q  += __shfl_xor(q,  off, 32);
                c  += __shfl_xor(c,  off, 32);
                qt += __shfl_xor(qt, off, 32);
            }
            float P = 1.f;
            for (int s = 0; s < nsteps; ++s) {
                const float alpha = __expf(-q * inv_den);
                const float beta  = h * alpha;
                const float om    = 1.0f - beta;
                const float qn    = om*om*q + 2.f*beta*om*c + beta*beta*qt;
                c = om*c + beta*qt;
                q = qn;
                P *= om;
            }
            const int rowInBlock = wave * ROWS_PER_WAVE + m * 16 + r + ((lane >> 4) << 3);
            if (rsel == 0) lds_P[rowInBlock] = P;   // lanes 0 and 16 write their rows
        }
    }
    __syncthreads();

    // ---- fused epilogue: out = P*x + (1-P)*target  (x tile is L2-hot) ----
    const size_t baseElem = (size_t)blockRow * D_MODEL;
    const v4f* __restrict__ xin = (const v4f*)(x + baseElem);
    const v4f* __restrict__ tv  = (const v4f*)tgt;
    v4f* __restrict__ o = (v4f*)(out + baseElem);

    for (int idx = tid; idx < ROWS_PER_BLOCK * COL4; idx += 256) {
        const int row = idx >> 10;          // / COL4
        const int c4  = idx & (COL4 - 1);
        const float P  = lds_P[row];
        const float Q  = 1.0f - P;
        const v4f xx = xin[idx];
        const v4f tt = tv[c4];
        v4f r;
        r.x = fmaf(P, xx.x, Q * tt.x);
        r.y = fmaf(P, xx.y, Q * tt.y);
        r.z = fmaf(P, xx.z, Q * tt.z);
        r.w = fmaf(P, xx.w, Q * tt.w);
        o[idx] = r;
    }
}

extern "C" void kernel_launch(void* const* d_in, const int* in_sizes, int n_in,
                              void* d_out, int out_size, void* d_ws, size_t ws_size,
                              hipStream_t stream) {
    (void)in_sizes; (void)n_in; (void)out_size; (void)d_ws; (void)ws_size;
    const float* x     = (const float*)d_in[0];
    const float* mu    = (const float*)d_in[1];
    const float* U     = (const float*)d_in[2];
    const float* S     = (const float*)d_in[3];
    const float* tgt   = (const float*)d_in[4];
    const float* lstep = (const float*)d_in[5];
    const float* sigma = (const float*)d_in[6];
    const int*   nstep = (const int*)d_in[7];
    float* out = (float*)d_out;

    const int BATCH = 32768;
    dim3 grid(BATCH / ROWS_PER_BLOCK);   // 128 blocks
    dim3 block(256);                      // 8 wave32
    gated_ode_flow_kernel<<<grid, block, 0, stream>>>(
        x, mu, U, S, tgt, lstep, sigma, nstep, out);
}